// TSACoding_70231305224552
// MI455X (gfx1250) — compile-verified
//
#include <hip/hip_runtime.h>
#include <hip/hip_bf16.h>
#include <math.h>

typedef __attribute__((ext_vector_type(2))) float v2f;
typedef __attribute__((ext_vector_type(8))) float v8f;

#define NI   64
#define NC   64
#define MAXR 36
#define MAXW 32
#define DIM  1024

__device__ __forceinline__ float wave_sum(float x) {
  #pragma unroll
  for (int off = 16; off > 0; off >>= 1) x += __shfl_xor(x, off, 32);
  return x;
}
__device__ __forceinline__ float wave_max(float x) {
  #pragma unroll
  for (int off = 16; off > 0; off >>= 1) x = fmaxf(x, __shfl_xor(x, off, 32));
  return x;
}

// ---------------------------------------------------------------------------
// Kernel A: per-caption Gram matrices of the *masked* captions.
//   Cg[c, w, w'] = sum_d caps_m[c,w,d] * caps_m[c,w',d]      (32x32 per c)
// ---------------------------------------------------------------------------
__global__ __launch_bounds__(256) void gram_kernel(const float* __restrict__ caps,
                                                   const int* __restrict__ cap_lens,
                                                   float* __restrict__ Cg) {
  const int c   = blockIdx.x;
  const int tid = threadIdx.x;
  const int cln = cap_lens[c];

  __shared__ float lds[32 * 129];          // 32 rows x 128 K-chunk, pad to 129
  float acc[4] = {0.f, 0.f, 0.f, 0.f};

  for (int kk = 0; kk < DIM; kk += 128) {
    __syncthreads();
    // stage masked chunk: 32*128 = 4096 floats, 16 per thread (coalesced)
    #pragma unroll
    for (int s = 0; s < 16; ++s) {
      int idx = tid + s * 256;             // 0..4095
      int w = idx >> 7, d = idx & 127;
      float v = 0.f;
      if (w < cln) v = caps[((size_t)c * MAXW + w) * DIM + kk + d];
      lds[w * 129 + d] = v;
    }
    __syncthreads();
    // each thread owns 4 of the 1024 outputs
    #pragma unroll
    for (int s = 0; s < 4; ++s) {
      int o = tid + s * 256;
      const float* ra = &lds[(o >> 5) * 129];
      const float* rb = &lds[(o & 31) * 129];
      float sum = acc[s];
      #pragma unroll 8
      for (int d = 0; d < 128; ++d) sum = fmaf(ra[d], rb[d], sum);
      acc[s] = sum;
    }
  }
  #pragma unroll
  for (int s = 0; s < 4; ++s)
    Cg[(size_t)c * 1024 + tid + s * 256] = acc[s];
}

// ---------------------------------------------------------------------------
// Kernel B: one (i,c) pair per block. 192 threads = 6 waves.
//   Phase 1: S_raw(48x32) = imgs[i](48x1024, zero-padded rows) @ caps[c]^T
//            via v_wmma_f32_16x16x4_f32, 3x2 tiles of 16x16, K staged in LDS.
//   Phase 2: per-row leaky_relu -> mask -> l2norm -> /T -> softmax -> mask,
//            num = sum_w p*raw, den = sqrt(p^T G_c p) - 1e-8, out or -1.
// ---------------------------------------------------------------------------
__global__ __launch_bounds__(192) void tsa_kernel(const float* __restrict__ imgs,
                                                  const float* __restrict__ caps,
                                                  const int* __restrict__ img_lens,
                                                  const int* __restrict__ cap_lens,
                                                  const float* __restrict__ Cg,
                                                  float* __restrict__ out) {
  const int c    = blockIdx.x;
  const int i    = blockIdx.y;
  const int tid  = threadIdx.x;
  const int lane = tid & 31;
  const int wave = tid >> 5;

  __shared__ float sA[48 * 36];   // 48 rows x 32 K, stride 36 (16B aligned, conflict-free)
  __shared__ float sB[32 * 36];   // 32 rows x 32 K, stride 36
  __shared__ float sS[48 * 33];   // S_raw, stride 33
  __shared__ float sC[32 * 33];   // Gram of caption c, stride 33

  // preload Gram for this caption (used in phase 2; later barriers cover it)
  for (int idx = tid; idx < 1024; idx += 192)
    sC[(idx >> 5) * 33 + (idx & 31)] = Cg[(size_t)c * 1024 + idx];

  const int r0   = (wave >> 1) * 16;   // row tile: 0,16,32
  const int w0   = (wave & 1) * 16;    // col tile: 0,16
  const int half = lane >> 4;          // 0: lanes 0-15, 1: lanes 16-31
  const int mrow = lane & 15;

  v8f acc = {0.f, 0.f, 0.f, 0.f, 0.f, 0.f, 0.f, 0.f};

  for (int kk = 0; kk < DIM; kk += 32) {
    __syncthreads();
    // A chunk: 48 x 32 = 384 float4, 2 per thread (rows >= 36 zero-filled)
    #pragma unroll
    for (int s = 0; s < 2; ++s) {
      int idx = tid + s * 192;
      int row = idx >> 3;
      int kq  = (idx & 7) << 2;
      float4 v = {0.f, 0.f, 0.f, 0.f};
      if (row < MAXR)
        v = *(const float4*)(imgs + ((size_t)i * MAXR + row) * DIM + kk + kq);
      *(float4*)(&sA[row * 36 + kq]) = v;
    }
    // B chunk: 32 x 32 = 256 float4
    #pragma unroll
    for (int s = 0; s < 2; ++s) {
      int idx = tid + s * 192;
      if (idx < 256) {
        int row = idx >> 3;
        int kq  = (idx & 7) << 2;
        *(float4*)(&sB[row * 36 + kq]) =
            *(const float4*)(caps + ((size_t)c * MAXW + row) * DIM + kk + kq);
      }
    }
    __syncthreads();
    // 8 WMMA steps over this K chunk (K granularity 4)
    #pragma unroll
    for (int ks = 0; ks < 32; ks += 4) {
      // A 16x4 layout: lanes 0-15 hold K=ks,ks+1 ; lanes 16-31 hold K=ks+2,ks+3
      v2f a = *(const v2f*)(&sA[(r0 + mrow) * 36 + ks + 2 * half]);
      // B 4x16 layout mirrors A over the N dimension
      v2f b = *(const v2f*)(&sB[(w0 + mrow) * 36 + ks + 2 * half]);
      acc = __builtin_amdgcn_wmma_f32_16x16x4_f32(
          /*neg_a=*/false, a, /*neg_b=*/false, b,
          /*c_mod=*/(short)0, acc, /*reuse_a=*/false, /*reuse_b=*/false);
    }
  }

  // C/D layout: VGPR j -> M=j (lanes 0-15) or M=j+8 (lanes 16-31), N=lane%16
  #pragma unroll
  for (int j = 0; j < 8; ++j)
    sS[(r0 + j + 8 * half) * 33 + (w0 + mrow)] = acc[j];
  __syncthreads();

  const int  iln = img_lens[i];
  const int  cln = cap_lens[c];
  const bool wv  = lane < cln;

  // 6 waves x 6 rows = 36 rows; lane <-> w position
  for (int r = wave; r < MAXR; r += 6) {
    float outv = -1.0f;
    if (r < iln) {
      float raw = sS[r * 33 + lane];
      float lr  = raw > 0.f ? raw : 0.1f * raw;   // leaky_relu(0.1)
      if (!wv) lr = 0.f;                          // fgmask -> 0
      float ss = wave_sum(lr * lr);
      float v  = lr * (10.0f / (sqrtf(ss) + 1e-8f));  // l2norm then /T (T=0.1)
      if (!wv) v = -1e30f;                        // fgmask -> NEG
      float m  = wave_max(v);
      float e  = wv ? __expf(v - m) : 0.f;
      float se = wave_sum(e);
      float p  = e / se;                          // softmax over w, masked to 0
      float num = wave_sum(p * raw);              // attended . imgs_m (algebraic)
      // t[lane] = sum_w' G[lane][w'] * p[w']  (broadcast p across lanes)
      float t = 0.f;
      const float* crow = &sC[lane * 33];
      #pragma unroll 8
      for (int wp = 0; wp < 32; ++wp) {
        float pb = __shfl(p, wp, 32);
        t = fmaf(crow[wp], pb, t);
      }
      float q   = wave_sum(p * t);                // p^T G p = ||attended||^2
      float den = sqrtf(q) + (-1e-8f);            // EPS is NEGATIVE in source
      outv = num / den;
    }
    if (lane == 0) out[((size_t)i * NC + c) * MAXR + r] = outv;
  }
}

extern "C" void kernel_launch(void* const* d_in, const int* in_sizes, int n_in,
                              void* d_out, int out_size, void* d_ws, size_t ws_size,
                              hipStream_t stream) {
  const float* imgs     = (const float*)d_in[0];
  const float* caps     = (const float*)d_in[1];
  const int*   img_lens = (const int*)d_in[2];
  const int*   cap_lens = (const int*)d_in[3];
  float*       out      = (float*)d_out;
  float*       Cg       = (float*)d_ws;   // 64*32*32 floats = 256 KB scratch

  gram_kernel<<<dim3(NC), 256, 0, stream>>>(caps, cap_lens, Cg);
  tsa_kernel<<<dim3(NC, NI), 192, 0, stream>>>(imgs, caps, img_lens, cap_lens, Cg, out);
}